// APSLayer_12524124635495
// MI455X (gfx1250) — compile-verified
//
#include <hip/hip_runtime.h>

typedef __attribute__((ext_vector_type(2))) float v2f;
typedef __attribute__((ext_vector_type(8))) float v8f;

#define B_   32
#define H_   128
#define W_   128
#define C_   128
#define HH_  64
#define WW_  64
#define NPOS (B_ * HH_ * WW_)      // 131072 positions per component
#define OUTN (NPOS * C_)           // 16777216 floats per component
#define NB1  512                   // pass-1 blocks per component
#define T1   256                   // pass-1 threads per block (8 waves)
#define WAVES1 (T1 / 32)
#define TOTWAVES (NB1 * WAVES1)    // 4096 waves per component
#define ITERS (NPOS / TOTWAVES)    // 32 positions per wave (== batch index)
#define BATCH_STRIDE (H_ * W_ * C_) // 2^21 floats between consecutive batches

// ---------------------------------------------------------------------------
// Pass 1: per-component partial sums of squares.
// grid = (NB1, 4). Wave gw (< 4096) owns one fixed (hh,ww) position and walks
// the batch dimension: p = gw + it*4096 implies b == it and hh/ww constant,
// so the address is base + it*8MB -> pure pointer increment in the loop.
// Lane L loads float4 at channel 4L: the wave covers one contiguous 512B
// chunk belonging entirely to component blockIdx.y. Squares feed the A
// operand of V_WMMA_F32_16X16X4_F32 with an all-ones B; the 16x16 f32 C tile
// is a 256-slot accumulator (tile total = 16 * true sum, layout-invariant).
// ---------------------------------------------------------------------------
__global__ __launch_bounds__(T1) void aps_pass1(const float* __restrict__ x,
                                                float* __restrict__ partial) {
  const int k = blockIdx.y;            // component 0..3
  const int i = k >> 1, j = k & 1;
  const int lane = threadIdx.x & 31;
  const int wave = threadIdx.x >> 5;
  const int gw = blockIdx.x * WAVES1 + wave;   // 0..4095
  const int hh = (gw >> 6) & 63;
  const int ww = gw & 63;

  const float* ptr = x + ((2 * hh + i) * W_ + (2 * ww + j)) * C_ + lane * 4;

  v8f acc = {};
  v2f ones; ones.x = 1.0f; ones.y = 1.0f;

  #pragma unroll 8
  for (int it = 0; it < ITERS; ++it) {
    const float4 v = *(const float4*)ptr;
    ptr += BATCH_STRIDE;

    v2f a0; a0.x = v.x * v.x; a0.y = v.y * v.y;
    v2f a1; a1.x = v.z * v.z; a1.y = v.w * v.w;
    // D = A*ones + C : accumulates the sum of all A entries into the tile.
    acc = __builtin_amdgcn_wmma_f32_16x16x4_f32(false, a0, false, ones,
                                                (short)0, acc, false, false);
    acc = __builtin_amdgcn_wmma_f32_16x16x4_f32(false, a1, false, ones,
                                                (short)0, acc, false, false);
  }

  // Tile total = 16 * (sum of squares seen by this wave).
  float s = acc[0] + acc[1] + acc[2] + acc[3] + acc[4] + acc[5] + acc[6] + acc[7];

  __shared__ float red[T1];
  red[threadIdx.x] = s;
  __syncthreads();
  for (int st = T1 / 2; st > 0; st >>= 1) {
    if (threadIdx.x < st) red[threadIdx.x] += red[threadIdx.x + st];
    __syncthreads();
  }
  if (threadIdx.x == 0)
    partial[k * NB1 + blockIdx.x] = red[0] * (1.0f / 16.0f);
}

// ---------------------------------------------------------------------------
// Pass 2: reduce NB1 partials per component, argmax (strict > keeps first
// max, matching jnp.argmax), write index to workspace (int) and to the tail
// of d_out (float value).
// ---------------------------------------------------------------------------
__global__ __launch_bounds__(256) void aps_pass2(const float* __restrict__ partial,
                                                 int* __restrict__ idx_out,
                                                 float* __restrict__ out_idx_f) {
  __shared__ float red[256];
  __shared__ float sums[4];
  for (int k = 0; k < 4; ++k) {
    red[threadIdx.x] = partial[k * NB1 + threadIdx.x] +
                       partial[k * NB1 + threadIdx.x + 256];
    __syncthreads();
    for (int st = 128; st > 0; st >>= 1) {
      if (threadIdx.x < st) red[threadIdx.x] += red[threadIdx.x + st];
      __syncthreads();
    }
    if (threadIdx.x == 0) sums[k] = red[0];
    __syncthreads();
  }
  if (threadIdx.x == 0) {
    int best = 0;
    float bv = sums[0];
    for (int k = 1; k < 4; ++k)
      if (sums[k] > bv) { bv = sums[k]; best = k; }
    *idx_out = best;
    *out_idx_f = (float)best;     // index as last element of d_out
  }
}

// ---------------------------------------------------------------------------
// Pass 3: coalesced float4 gather-copy of the winning component.
// ---------------------------------------------------------------------------
__global__ __launch_bounds__(256) void aps_pass3(const float* __restrict__ x,
                                                 const int* __restrict__ idx_in,
                                                 float* __restrict__ out) {
  const int k = *idx_in;               // wave-uniform
  const int i = k >> 1, j = k & 1;
  const int q  = blockIdx.x * blockDim.x + threadIdx.x;  // float4 index
  const int c4 = q & 31;               // 32 float4 per position (C=128)
  const int p  = q >> 5;
  const int b  = p >> 12;
  const int hh = (p >> 6) & 63;
  const int ww = p & 63;
  const int src = ((b * H_ + 2 * hh + i) * W_ + (2 * ww + j)) * C_ + c4 * 4;
  *(float4*)(out + (size_t)q * 4) = *(const float4*)(x + src);
}

extern "C" void kernel_launch(void* const* d_in, const int* in_sizes, int n_in,
                              void* d_out, int out_size, void* d_ws, size_t ws_size,
                              hipStream_t stream) {
  (void)in_sizes; (void)n_in; (void)out_size; (void)ws_size;
  const float* x = (const float*)d_in[0];
  float* out = (float*)d_out;
  float* partial = (float*)d_ws;                       // 4*NB1 floats = 8 KB
  int* idxp = (int*)((char*)d_ws + 4 * NB1 * sizeof(float));

  dim3 g1(NB1, 4, 1);
  aps_pass1<<<g1, T1, 0, stream>>>(x, partial);
  aps_pass2<<<1, 256, 0, stream>>>(partial, idxp, out + OUTN);
  aps_pass3<<<OUTN / 1024, 256, 0, stream>>>(x, idxp, out);
}